// CAM_Module_56444460204534
// MI455X (gfx1250) — compile-verified
//
#include <hip/hip_runtime.h>
#include <hip/hip_bf16.h>

// CAM module (DANet channel attention) for MI455X / gfx1250, fp32 throughout.
// Memory-bound problem (~384 MiB min traffic @ 23.3 TB/s ~= 16us); matmuls use
// V_WMMA_F32_16X16X4_F32. BN stats derived algebraically from the Gram matrix
// (E[out^2] = a^T E a) to avoid a third pass over x.

typedef float v2f __attribute__((ext_vector_type(2)));
typedef float v8f __attribute__((ext_vector_type(8)));

#define BB 16
#define CC 128
#define NN 16384            // H*W
#define SPLIT 16            // K-splits for the Gram reduction
#define KS (NN / SPLIT)     // 1024

__device__ __forceinline__ v8f wmma_f32(v2f a, v2f b, v8f c) {
  // D = A(16x4) * B(4x16) + C(16x16), fp32
  return __builtin_amdgcn_wmma_f32_16x16x4_f32(
      /*neg_a=*/false, a, /*neg_b=*/false, b,
      /*c_mod=*/(short)0, c, /*reuse_a=*/false, /*reuse_b=*/false);
}

// ---------------------------------------------------------------------------
// K1: partial Gram matrices  pE[b][s] += Q_b[:, ks] Q_b[:, ks]^T
//     + partial per-channel sums pS (for BN mean).
// Grid: BB*SPLIT blocks x 256 threads (8 waves). Wave w: row tile w, 8 col tiles.
// ---------------------------------------------------------------------------
__global__ void __launch_bounds__(256) gram_partial_kernel(
    const float* __restrict__ x, float* __restrict__ pE, float* __restrict__ pS) {
  const int blk = blockIdx.x;
  const int b = blk >> 4;
  const int s = blk & 15;
  const int k0 = s * KS;
  const int tid = threadIdx.x;
  const int lane = tid & 31;
  const int wv = tid >> 5;
  const int lm = lane & 15;
  const int half = lane >> 4;
  const float* xb = x + (size_t)b * CC * NN;

  // Per-channel partial sums: 2 halves per channel, all 256 threads busy.
  {
    const int ch = tid & 127;
    const int part = tid >> 7;
    const float* p = xb + (size_t)ch * NN + k0 + part * (KS / 2);
    float sum = 0.f;
#pragma unroll 8
    for (int k = 0; k < KS / 2; ++k) sum += p[k];
    pS[((size_t)(b * SPLIT + s) * 2 + part) * CC + ch] = sum;
  }

  // A frag: A[m][k], m = lane%16 (row = 16*wv + m), k = 2*half + e  -> v2f load
  // B frag: B[k][n], n = lane%16 (col = 16*j  + n), k = 2*half + e  -> v2f load
  v8f acc[8] = {};
  const float* aptr = xb + (size_t)(16 * wv + lm) * NN + k0 + 2 * half;
  const float* bptr[8];
#pragma unroll
  for (int j = 0; j < 8; ++j)
    bptr[j] = xb + (size_t)(16 * j + lm) * NN + k0 + 2 * half;

  for (int kk = 0; kk < KS; kk += 4) {
    __builtin_prefetch(aptr + kk + 128, 0, 0);
    v2f af = *(const v2f*)(aptr + kk);
#pragma unroll
    for (int j = 0; j < 8; ++j) {
      v2f bf = *(const v2f*)(bptr[j] + kk);
      acc[j] = wmma_f32(af, bf, acc[j]);
    }
  }

  // C/D layout: vgpr r, lane l -> M = r + 8*(l/16), N = l%16
  float* dst = pE + (size_t)(b * SPLIT + s) * (CC * CC);
#pragma unroll
  for (int j = 0; j < 8; ++j) {
#pragma unroll
    for (int r = 0; r < 8; ++r) {
      const int row = 16 * wv + 8 * half + r;
      const int col = 16 * j + lm;
      dst[row * CC + col] = acc[j][r];
    }
  }
}

// ---------------------------------------------------------------------------
// K2: per batch: reduce partials -> E (LDS), softmax(max-e) == exp(min-e)/sum,
// fold gamma into attention rows; emit m1[b,c] = sum_j a_j s_j and
// e2[b,c] = a^T E a (second moment of out via Gram identity).
// Grid: BB blocks x 256 threads, dynamic LDS = (CC*CC + CC)*4 bytes.
// ---------------------------------------------------------------------------
__global__ void __launch_bounds__(256) softmax_stats_kernel(
    const float* __restrict__ pE, const float* __restrict__ pS,
    const float* __restrict__ gamma_p, float* __restrict__ attn,
    float* __restrict__ m1, float* __restrict__ e2) {
  extern __shared__ float smem[];
  float* E = smem;              // CC*CC
  float* sv = smem + CC * CC;   // CC
  const int b = blockIdx.x;
  const int tid = threadIdx.x;

  for (int idx = tid; idx < CC * CC; idx += 256) {
    float v = 0.f;
#pragma unroll
    for (int s = 0; s < SPLIT; ++s)
      v += pE[(size_t)(b * SPLIT + s) * (CC * CC) + idx];
    E[idx] = v;
  }
  if (tid < CC) {
    float v = 0.f;
#pragma unroll
    for (int s = 0; s < 2 * SPLIT; ++s)
      v += pS[(size_t)(b * 2 * SPLIT + s) * CC + tid];
    sv[tid] = v;
  }
  __syncthreads();

  if (tid < CC) {
    const float g = gamma_p[0];
    const float* Er = E + tid * CC;
    float mn = Er[0];
    for (int j = 1; j < CC; ++j) mn = fminf(mn, Er[j]);
    float sum = 0.f;
    for (int j = 0; j < CC; ++j) sum += __expf(mn - Er[j]);
    const float inv = g / sum;   // gamma folded into attention

    float a[CC];
    float mm = 0.f;
    for (int j = 0; j < CC; ++j) {
      const float aj = __expf(mn - Er[j]) * inv;
      a[j] = aj;
      mm += aj * sv[j];
      attn[(size_t)b * CC * CC + tid * CC + j] = aj;
    }
    m1[b * CC + tid] = mm;

    float acc = 0.f;
    for (int j = 0; j < CC; ++j) {
      const float* Ej = E + j * CC;
      float wj = 0.f;
      for (int k = 0; k < CC; ++k) wj += Ej[k] * a[k];
      acc += a[j] * wj;
    }
    e2[b * CC + tid] = acc;      // includes gamma^2 via folded rows
  }
}

// ---------------------------------------------------------------------------
// K3: mean/scale per channel. Grid: 1 block x 128 threads.
// ---------------------------------------------------------------------------
__global__ void finalize_stats_kernel(
    const float* __restrict__ m1, const float* __restrict__ e2,
    const float* __restrict__ bn_w, float* __restrict__ meanv,
    float* __restrict__ scalev) {
  const int c = threadIdx.x;
  if (c >= CC) return;
  const float invBN = 1.0f / ((float)BB * (float)NN);
  float mm = 0.f, ee = 0.f;
#pragma unroll
  for (int b = 0; b < BB; ++b) {
    mm += m1[b * CC + c];
    ee += e2[b * CC + c];
  }
  mm *= invBN;
  ee *= invBN;
  const float var = ee - mm * mm;   // biased variance
  meanv[c] = mm;
  scalev[c] = bn_w[c] * rsqrtf(var + 1e-5f);
}

// ---------------------------------------------------------------------------
// K4: y = (A_g x - mean) * scale + bias + x, fused epilogue.
// Grid: BB * (NN/128) blocks x 256 threads. Wave w owns col tile w (16 n's),
// all 8 row tiles -> x is read exactly once per element (no cross-wave dup).
// ---------------------------------------------------------------------------
__global__ void __launch_bounds__(256) attn_apply_kernel(
    const float* __restrict__ x, const float* __restrict__ attn,
    const float* __restrict__ meanv, const float* __restrict__ scalev,
    const float* __restrict__ bn_b, float* __restrict__ out) {
  __shared__ float sm[3 * CC];
  const int b = blockIdx.x >> 7;
  const int chunk = blockIdx.x & 127;
  const int tid = threadIdx.x;
  if (tid < CC) {
    sm[tid] = meanv[tid];
    sm[CC + tid] = scalev[tid];
    sm[2 * CC + tid] = bn_b[tid];
  }
  __syncthreads();

  const int lane = tid & 31;
  const int wv = tid >> 5;
  const int lm = lane & 15;
  const int half = lane >> 4;
  const int n = chunk * 128 + 16 * wv + lm;
  const float* xb = x + (size_t)b * CC * NN;
  const float* ab = attn + (size_t)b * CC * CC;

  v8f acc[8] = {};
  for (int kk = 0; kk < CC; kk += 4) {
    const int kidx = kk + 2 * half;
    v2f bf;
    bf.x = xb[(size_t)kidx * NN + n];        // B[k][n] = x[b][k][n]
    bf.y = xb[(size_t)(kidx + 1) * NN + n];
#pragma unroll
    for (int i = 0; i < 8; ++i) {
      v2f af = *(const v2f*)(ab + (16 * i + lm) * CC + kidx);  // contiguous k
      acc[i] = wmma_f32(af, bf, acc[i]);
    }
  }

#pragma unroll
  for (int i = 0; i < 8; ++i) {
#pragma unroll
    for (int r = 0; r < 8; ++r) {
      const int c = 16 * i + 8 * half + r;
      const size_t idx = (size_t)c * NN + n;
      out[(size_t)b * CC * NN + idx] =
          (acc[i][r] - sm[c]) * sm[CC + c] + sm[2 * CC + c] + xb[idx];
    }
  }
}

// ---------------------------------------------------------------------------
extern "C" void kernel_launch(void* const* d_in, const int* in_sizes, int n_in,
                              void* d_out, int out_size, void* d_ws, size_t ws_size,
                              hipStream_t stream) {
  const float* x     = (const float*)d_in[0];
  const float* gamma = (const float*)d_in[1];
  const float* bn_w  = (const float*)d_in[2];
  const float* bn_b  = (const float*)d_in[3];
  float* out = (float*)d_out;

  // Workspace layout (floats). Total ~4.53M floats (~17.3 MiB).
  float* ws    = (float*)d_ws;
  float* pE    = ws;                                   // BB*SPLIT*CC*CC = 4,194,304
  float* pS    = pE + (size_t)BB * SPLIT * CC * CC;    // BB*SPLIT*2*CC  = 65,536
  float* attn  = pS + (size_t)BB * SPLIT * 2 * CC;     // BB*CC*CC       = 262,144
  float* m1    = attn + (size_t)BB * CC * CC;          // BB*CC          = 2,048
  float* e2    = m1 + BB * CC;                         // BB*CC          = 2,048
  float* meanv = e2 + BB * CC;                         // CC
  float* scalev = meanv + CC;                          // CC

  gram_partial_kernel<<<BB * SPLIT, 256, 0, stream>>>(x, pE, pS);
  softmax_stats_kernel<<<BB, 256, (CC * CC + CC) * sizeof(float), stream>>>(
      pE, pS, gamma, attn, m1, e2);
  finalize_stats_kernel<<<1, 128, 0, stream>>>(m1, e2, bn_w, meanv, scalev);
  attn_apply_kernel<<<BB * (NN / 128), 256, 0, stream>>>(
      x, attn, meanv, scalev, bn_b, out);
}